// YinYanGNN_27066883899969
// MI455X (gfx1250) — compile-verified
//
#include <hip/hip_runtime.h>
#include <hip/hip_bf16.h>

// ---------------------------------------------------------------------------
// YinYanGNN for MI455X (gfx1250, wave32)
//   GEMMs: V_WMMA_F32_16X16X4_F32 (fp32 WMMA -- GEMM cost is negligible vs the
//          L2-atomic-bound SpMM propagation, so keep full precision)
//   SpMM:  1 wave per edge, lane = feature; coalesced 128B gather + 128B
//          float-atomic scatter. All hot node arrays (12.8MB each) are
//          L2-resident on the 192MB L2.
// ---------------------------------------------------------------------------

#define NNODES 100000
#define NEDGES 3200000
#define INFEATS 128
#define HID 32
#define PROP 8

typedef __attribute__((ext_vector_type(2))) float v2f;
typedef __attribute__((ext_vector_type(8))) float v8f;

// ---------------- utility kernels ----------------

__global__ void zero_kernel(float* __restrict__ p, int n) {
    int i = blockIdx.x * blockDim.x + threadIdx.x;
    if (i < n) p[i] = 0.0f;
}

// deg layout: [0,N) src_pos | [N,2N) dst_pos | [2N,3N) src_neg | [3N,4N) dst_neg
__global__ void degree_kernel(const int* __restrict__ src, const int* __restrict__ dst,
                              const int* __restrict__ nsrc, const int* __restrict__ ndst,
                              float* __restrict__ deg) {
    int i = blockIdx.x * blockDim.x + threadIdx.x;
    if (i < NEDGES) {
        atomicAdd(&deg[src[i]], 1.0f);
        atomicAdd(&deg[NNODES + dst[i]], 1.0f);
    } else if (i < 2 * NEDGES) {
        int e = i - NEDGES;
        atomicAdd(&deg[2 * NNODES + nsrc[e]], 1.0f);
        atomicAdd(&deg[3 * NNODES + ndst[e]], 1.0f);
    }
}

__global__ void rsqrt_kernel(float* __restrict__ deg) {
    int i = blockIdx.x * blockDim.x + threadIdx.x;
    if (i < 4 * NNODES) deg[i] = rsqrtf(fmaxf(deg[i], 1.0f));
}

// ---------------- GEMM1: x = in_feat @ W1 + b1  (WMMA f32 16x16x4) ----------
// grid = (NNODES/16, HID/16), block = 32 (one wave per 16x16 output tile)
__global__ void gemm1_wmma_kernel(const float* __restrict__ A,
                                  const float* __restrict__ W1,
                                  const float* __restrict__ b1,
                                  float* __restrict__ X) {
    const int lane = threadIdx.x;          // 0..31, wave32
    const int row0 = blockIdx.x * 16;
    const int col0 = blockIdx.y * 16;
    const int half = lane >> 4;            // 0 or 1
    const int l    = lane & 15;
    const float* arow = A + (size_t)(row0 + l) * INFEATS;

    v8f c = {};
    for (int k0 = 0; k0 < INFEATS; k0 += 4) {
        const int ka = k0 + 2 * half;      // K indices per ISA 16x4 f32 A layout
        v2f a, b;
        a.x = arow[ka];
        a.y = arow[ka + 1];
        b.x = W1[(size_t)(ka)     * HID + col0 + l];
        b.y = W1[(size_t)(ka + 1) * HID + col0 + l];
        c = __builtin_amdgcn_wmma_f32_16x16x4_f32(false, a, false, b,
                                                  (short)0, c, false, false);
    }
    const float bias = b1[col0 + l];
#pragma unroll
    for (int v = 0; v < 8; ++v) {
        int m = v + 8 * half;              // C layout: VGPR v -> M = v (+8 upper half)
        X[(size_t)(row0 + m) * HID + col0 + l] = c[v] + bias;
    }
}

// ---------------- BatchNorm batch statistics -------------------------------

__global__ void bn_reduce_kernel(const float* __restrict__ X,
                                 float* __restrict__ colsum,
                                 float* __restrict__ colsq) {
    int tid = blockIdx.x * blockDim.x + threadIdx.x;
    int col = tid & (HID - 1);
    int row = tid >> 5;
    int rstride = (gridDim.x * blockDim.x) >> 5;
    float s = 0.0f, q = 0.0f;
    for (int r = row; r < NNODES; r += rstride) {
        float v = X[(size_t)r * HID + col];
        s += v;
        q += v * v;
    }
    atomicAdd(&colsum[col], s);
    atomicAdd(&colsq[col], q);
}

__global__ void bn_params_kernel(const float* __restrict__ colsum,
                                 const float* __restrict__ colsq,
                                 const float* __restrict__ gamma,
                                 const float* __restrict__ beta,
                                 float* __restrict__ scale,
                                 float* __restrict__ shift) {
    int c = threadIdx.x;
    if (c < HID) {
        float mu  = colsum[c] * (1.0f / NNODES);
        float var = colsq[c] * (1.0f / NNODES) - mu * mu;   // population variance
        float inv = rsqrtf(var + 1e-5f);
        float sc  = gamma[c] * inv;
        scale[c] = sc;
        shift[c] = beta[c] - mu * sc;
    }
}

// -------- GEMM2: ori_h = relu(bn(x)) @ W2 + b2; also h = ori_h (WMMA) ------
__global__ void gemm2_wmma_kernel(const float* __restrict__ X,
                                  const float* __restrict__ scale,
                                  const float* __restrict__ shift,
                                  const float* __restrict__ W2,
                                  const float* __restrict__ b2,
                                  float* __restrict__ ori_h,
                                  float* __restrict__ h) {
    const int lane = threadIdx.x;
    const int row0 = blockIdx.x * 16;
    const int col0 = blockIdx.y * 16;
    const int half = lane >> 4;
    const int l    = lane & 15;
    const float* xr = X + (size_t)(row0 + l) * HID;

    v8f c = {};
#pragma unroll
    for (int k0 = 0; k0 < HID; k0 += 4) {
        const int ka = k0 + 2 * half;
        v2f a, b;
        float a0 = xr[ka]     * scale[ka]     + shift[ka];
        float a1 = xr[ka + 1] * scale[ka + 1] + shift[ka + 1];
        a.x = fmaxf(a0, 0.0f);                 // ReLU fused into A-load
        a.y = fmaxf(a1, 0.0f);
        b.x = W2[(size_t)(ka)     * HID + col0 + l];
        b.y = W2[(size_t)(ka + 1) * HID + col0 + l];
        c = __builtin_amdgcn_wmma_f32_16x16x4_f32(false, a, false, b,
                                                  (short)0, c, false, false);
    }
    const float bias = b2[col0 + l];
#pragma unroll
    for (int v = 0; v < 8; ++v) {
        int m = v + 8 * half;
        float val = c[v] + bias;
        size_t idx = (size_t)(row0 + m) * HID + col0 + l;
        ori_h[idx] = val;
        h[idx] = val;
    }
}

// ---------------- propagation step kernels ---------------------------------

// hs_pos = h * rsqrt_src_pos[n]; hs_neg = h * rsqrt_src_neg[n]; zero aggs
__global__ void prop_prep_kernel(const float* __restrict__ h,
                                 const float* __restrict__ rs,
                                 float* __restrict__ hs_pos, float* __restrict__ hs_neg,
                                 float* __restrict__ agg_pos, float* __restrict__ agg_neg) {
    int i = blockIdx.x * blockDim.x + threadIdx.x;
    if (i < NNODES * HID) {
        int n = i >> 5;                     // HID == 32
        float hv = h[i];
        hs_pos[i] = hv * rs[n];
        hs_neg[i] = hv * rs[2 * NNODES + n];
        agg_pos[i] = 0.0f;
        agg_neg[i] = 0.0f;
    }
}

// one wave per edge (pos graph for wave < E, neg graph otherwise); lane = feature
__global__ void edge_scatter_kernel(const int* __restrict__ src, const int* __restrict__ dst,
                                    const int* __restrict__ nsrc, const int* __restrict__ ndst,
                                    const float* __restrict__ hs_pos,
                                    const float* __restrict__ hs_neg,
                                    float* __restrict__ agg_pos,
                                    float* __restrict__ agg_neg) {
    int wave = blockIdx.x * (blockDim.x >> 5) + (threadIdx.x >> 5);
    int lane = threadIdx.x & 31;
    if (wave < NEDGES) {
        int s = src[wave], d = dst[wave];
        atomicAdd(&agg_pos[(size_t)d * HID + lane], hs_pos[(size_t)s * HID + lane]);
    } else if (wave < 2 * NEDGES) {
        int e = wave - NEDGES;
        int s = nsrc[e], d = ndst[e];
        atomicAdd(&agg_neg[(size_t)d * HID + lane], hs_neg[(size_t)s * HID + lane]);
    }
}

// h = ori_h + agg_pos * rsqrt_dst_pos[n] - agg_neg * rsqrt_dst_neg[n]
__global__ void prop_finalize_kernel(const float* __restrict__ ori_h,
                                     const float* __restrict__ rs,
                                     const float* __restrict__ agg_pos,
                                     const float* __restrict__ agg_neg,
                                     float* __restrict__ h) {
    int i = blockIdx.x * blockDim.x + threadIdx.x;
    if (i < NNODES * HID) {
        int n = i >> 5;
        h[i] = ori_h[i] + agg_pos[i] * rs[NNODES + n] - agg_neg[i] * rs[3 * NNODES + n];
    }
}

// ---------------- launch ----------------------------------------------------

extern "C" void kernel_launch(void* const* d_in, const int* in_sizes, int n_in,
                              void* d_out, int out_size, void* d_ws, size_t ws_size,
                              hipStream_t stream) {
    const float* in_feat = (const float*)d_in[0];
    const float* W1      = (const float*)d_in[1];
    const float* b1      = (const float*)d_in[2];
    const float* gamma   = (const float*)d_in[3];
    const float* beta    = (const float*)d_in[4];
    const float* W2      = (const float*)d_in[5];
    const float* b2      = (const float*)d_in[6];
    const int*   src     = (const int*)d_in[7];
    const int*   dst     = (const int*)d_in[8];
    const int*   nsrc    = (const int*)d_in[9];
    const int*   ndst    = (const int*)d_in[10];

    float* h  = (float*)d_out;                  // [N, HID] final output, used as h
    float* ws = (float*)d_ws;

    const size_t NH = (size_t)NNODES * HID;
    float* x       = ws;                        // N*32
    float* ori_h   = x + NH;                    // N*32
    float* hs_pos  = ori_h + NH;                // N*32
    float* hs_neg  = hs_pos + NH;               // N*32
    float* agg_pos = hs_neg + NH;               // N*32
    float* agg_neg = agg_pos + NH;              // N*32
    float* deg     = agg_neg + NH;              // 4*N (becomes rsqrt normalizers)
    float* colsum  = deg + 4 * NNODES;          // 32
    float* colsq   = colsum + HID;              // 32
    float* scale   = colsq + HID;               // 32
    float* shift   = scale + HID;               // 32

    // 1) zero degree counters + BN accumulators (contiguous region)
    int zn = 4 * NNODES + 2 * HID;
    zero_kernel<<<(zn + 255) / 256, 256, 0, stream>>>(deg, zn);

    // 2) degrees (float atomics), then fold clip+rsqrt in place
    degree_kernel<<<(2 * NEDGES + 255) / 256, 256, 0, stream>>>(src, dst, nsrc, ndst, deg);
    rsqrt_kernel<<<(4 * NNODES + 255) / 256, 256, 0, stream>>>(deg);

    // 3) GEMM1 via WMMA: x = in_feat @ W1 + b1   (N = 6250*16 exactly)
    gemm1_wmma_kernel<<<dim3(NNODES / 16, HID / 16), 32, 0, stream>>>(in_feat, W1, b1, x);

    // 4) BatchNorm batch stats -> fused scale/shift
    bn_reduce_kernel<<<512, 256, 0, stream>>>(x, colsum, colsq);
    bn_params_kernel<<<1, 32, 0, stream>>>(colsum, colsq, gamma, beta, scale, shift);

    // 5) GEMM2 via WMMA: ori_h = relu(bn(x)) @ W2 + b2 ; h = ori_h
    gemm2_wmma_kernel<<<dim3(NNODES / 16, HID / 16), 32, 0, stream>>>(
        x, scale, shift, W2, b2, ori_h, h);

    // 6) 8 propagation steps: h = conv_pos(h) - conv_neg(h) + ori_h
    const int nNH  = (int)NH;
    const int gNH  = (nNH + 255) / 256;
    const int gEdg = (2 * NEDGES + 7) / 8;      // 8 waves (edges) per 256-thread block
    for (int step = 0; step < PROP; ++step) {
        prop_prep_kernel<<<gNH, 256, 0, stream>>>(h, deg, hs_pos, hs_neg, agg_pos, agg_neg);
        edge_scatter_kernel<<<gEdg, 256, 0, stream>>>(src, dst, nsrc, ndst,
                                                      hs_pos, hs_neg, agg_pos, agg_neg);
        prop_finalize_kernel<<<gNH, 256, 0, stream>>>(ori_h, deg, agg_pos, agg_neg, h);
    }
}